// SAGE_40956808134940
// MI455X (gfx1250) — compile-verified
//
#include <hip/hip_runtime.h>

typedef __attribute__((ext_vector_type(2))) float v2f;
typedef __attribute__((ext_vector_type(8))) float v8f;

#define N_NODES 50000
#define N_EDGES 800000
#define IN_CH   128
#define HID_CH  256
#define OUT_CH  128

// ---------------- utility kernels ----------------

__global__ __launch_bounds__(256) void zero_f32_vec4(float* __restrict__ p, int n4) {
    int i = blockIdx.x * blockDim.x + threadIdx.x;
    if (i < n4) {
        float4 z; z.x = 0.f; z.y = 0.f; z.z = 0.f; z.w = 0.f;
        ((float4*)p)[i] = z;
    }
}

__global__ __launch_bounds__(256) void deg_kernel(const int* __restrict__ dst,
                                                  float* __restrict__ deg, int E) {
    int e = blockIdx.x * blockDim.x + threadIdx.x;
    if (e < E) atomicAdd(&deg[dst[e]], 1.0f);
}

__global__ __launch_bounds__(256) void invdeg_kernel(const float* __restrict__ deg,
                                                     float* __restrict__ invdeg, int n) {
    int i = blockIdx.x * blockDim.x + threadIdx.x;
    if (i < n) invdeg[i] = 1.0f / fmaxf(deg[i], 1.0f);
}

// one thread = (edge, 4-channel group): float4 gather + 4 fp32 L2 atomics
template <int C>
__global__ __launch_bounds__(256) void scatter_kernel(const float* __restrict__ feat,
                                                      const int* __restrict__ src,
                                                      const int* __restrict__ dst,
                                                      float* __restrict__ agg, int E) {
    const int GR = C / 4;
    int gid = blockIdx.x * blockDim.x + threadIdx.x;
    if (gid >= E * GR) return;
    int e = gid / GR;
    int g = gid - e * GR;
    int s = src[e];
    int d = dst[e];
    float4 v = *(const float4*)(feat + (size_t)s * C + g * 4);
    float* p = agg + (size_t)d * C + g * 4;
    atomicAdd(p + 0, v.x);
    atomicAdd(p + 1, v.y);
    atomicAdd(p + 2, v.z);
    atomicAdd(p + 3, v.w);
}

// agg[row][*] *= invdeg[row]  (mean aggregation)
template <int C>
__global__ __launch_bounds__(256) void scale_rows(float* __restrict__ agg,
                                                  const float* __restrict__ invdeg, int total4) {
    int i = blockIdx.x * blockDim.x + threadIdx.x;
    if (i >= total4) return;
    int row = i / (C / 4);
    float s = invdeg[row];
    float4 v = ((float4*)agg)[i];
    v.x *= s; v.y *= s; v.z *= s; v.w *= s;
    ((float4*)agg)[i] = v;
}

// ---------------- WMMA fp32 GEMM ----------------
// out[M, NOUT] = A0[M,CIN] @ B0[NOUT,CIN]^T + A1[M,CIN] @ B1[NOUT,CIN]^T + bias, opt ReLU.
// One wave per 16x16 output tile; K accumulated 4 at a time with
// V_WMMA_F32_16X16X4_F32.
//
// Fragment layouts per CDNA5 ISA (CDNA5 ISA):
//   A 16x4 f32 : lanes 0-15 hold (K=k, K=k+1) for row M=lane; lanes 16-31 hold (K=k+2, K=k+3).
//   B 4x16 f32 : lanes 0-15 hold (K=k, K=k+1) for col N=lane&15; lanes 16-31 (K=k+2, K=k+3).
//   C/D 16x16  : VGPR r -> rows r (lanes 0-15) and r+8 (lanes 16-31), col = lane&15.
template <int CIN, int NOUT, bool RELU>
__global__ __launch_bounds__(256) void sage_gemm_wmma(const float* __restrict__ A0,
                                                      const float* __restrict__ A1,
                                                      const float* __restrict__ B0,
                                                      const float* __restrict__ B1,
                                                      const float* __restrict__ bias,
                                                      float* __restrict__ out, int M) {
    const int lane  = threadIdx.x & 31;
    const int wave  = (blockIdx.x * blockDim.x + threadIdx.x) >> 5;
    const int ntile = NOUT / 16;
    const int tm    = wave / ntile;
    const int tn    = wave - tm * ntile;
    if (tm * 16 >= M) return;              // wave-uniform: EXEC stays all-ones

    const int l15   = lane & 15;
    const int khalf = lane >> 4;           // 0: K=k,k+1   1: K=k+2,k+3
    const int row   = tm * 16 + l15;
    const int col   = tn * 16 + l15;

    const float* a0 = A0 + (size_t)row * CIN;
    const float* a1 = A1 + (size_t)row * CIN;
    const float* b0 = B0 + (size_t)col * CIN;   // weights are [NOUT, CIN] row-major
    const float* b1 = B1 + (size_t)col * CIN;

    v8f c = {};
#pragma unroll 8
    for (int k = 0; k < CIN; k += 4) {
        int ka = k + khalf * 2;
        v2f a, b;
        a.x = a0[ka]; a.y = a0[ka + 1];
        b.x = b0[ka]; b.y = b0[ka + 1];
        c = __builtin_amdgcn_wmma_f32_16x16x4_f32(false, a, false, b, (short)0, c, false, false);
    }
#pragma unroll 8
    for (int k = 0; k < CIN; k += 4) {
        int ka = k + khalf * 2;
        v2f a, b;
        a.x = a1[ka]; a.y = a1[ka + 1];
        b.x = b1[ka]; b.y = b1[ka + 1];
        c = __builtin_amdgcn_wmma_f32_16x16x4_f32(false, a, false, b, (short)0, c, false, false);
    }

    float bv = bias[col];
#pragma unroll
    for (int r = 0; r < 8; ++r) {
        int orow = tm * 16 + r + khalf * 8;
        float v  = c[r] + bv;
        if (RELU) v = fmaxf(v, 0.0f);
        out[(size_t)orow * NOUT + col] = v;
    }
}

// ---------------- launch ----------------

static inline int cdiv(int a, int b) { return (a + b - 1) / b; }

extern "C" void kernel_launch(void* const* d_in, const int* in_sizes, int n_in,
                              void* d_out, int out_size, void* d_ws, size_t ws_size,
                              hipStream_t stream) {
    const float* x   = (const float*)d_in[0];
    const int*   ei  = (const int*)d_in[1];
    const int*   src = ei;
    const int*   dst = ei + N_EDGES;
    const float* W1l = (const float*)d_in[2];
    const float* b1l = (const float*)d_in[3];
    const float* W1r = (const float*)d_in[4];
    const float* W2l = (const float*)d_in[5];
    const float* b2l = (const float*)d_in[6];
    const float* W2r = (const float*)d_in[7];
    float*       out = (float*)d_out;

    // workspace layout (floats):
    //   [0, N)                deg
    //   [N, 2N)               invdeg
    //   [2N, 2N + N*256)      h   (layer-1 output)
    //   [.., .. + N*256)      agg (layer-1 agg in first half, reused as layer-2 agg)
    float* ws     = (float*)d_ws;
    float* deg    = ws;
    float* invdeg = ws + N_NODES;
    float* h      = ws + 2 * (size_t)N_NODES;
    float* agg    = h + (size_t)N_NODES * HID_CH;

    const int B = 256;

    // ---- layer 1: agg1 = mean_{src->dst} x ----
    zero_f32_vec4<<<cdiv(N_NODES / 4, B), B, 0, stream>>>(deg, N_NODES / 4);
    zero_f32_vec4<<<cdiv(N_NODES * IN_CH / 4, B), B, 0, stream>>>(agg, N_NODES * IN_CH / 4);
    deg_kernel<<<cdiv(N_EDGES, B), B, 0, stream>>>(dst, deg, N_EDGES);
    scatter_kernel<IN_CH><<<cdiv(N_EDGES * (IN_CH / 4), B), B, 0, stream>>>(x, src, dst, agg, N_EDGES);
    invdeg_kernel<<<cdiv(N_NODES, B), B, 0, stream>>>(deg, invdeg, N_NODES);
    scale_rows<IN_CH><<<cdiv(N_NODES * IN_CH / 4, B), B, 0, stream>>>(agg, invdeg, N_NODES * IN_CH / 4);

    // h = relu(agg @ W1l^T + b1l + x @ W1r^T)   [50000 x 256]
    {
        int waves = (N_NODES / 16) * (HID_CH / 16);
        sage_gemm_wmma<IN_CH, HID_CH, true>
            <<<cdiv(waves * 32, B), B, 0, stream>>>(agg, x, W1l, W1r, b1l, h, N_NODES);
    }

    // ---- layer 2: agg2 = mean_{src->dst} h ----
    zero_f32_vec4<<<cdiv(N_NODES * HID_CH / 4, B), B, 0, stream>>>(agg, N_NODES * HID_CH / 4);
    scatter_kernel<HID_CH><<<cdiv(N_EDGES * (HID_CH / 4), B), B, 0, stream>>>(h, src, dst, agg, N_EDGES);
    scale_rows<HID_CH><<<cdiv(N_NODES * HID_CH / 4, B), B, 0, stream>>>(agg, invdeg, N_NODES * HID_CH / 4);

    // out = agg @ W2l^T + b2l + h @ W2r^T   [50000 x 128]
    {
        int waves = (N_NODES / 16) * (OUT_CH / 16);
        sage_gemm_wmma<HID_CH, OUT_CH, false>
            <<<cdiv(waves * 32, B), B, 0, stream>>>(agg, h, W2l, W2r, b2l, out, N_NODES);
    }
}